// GCN_37649683317295
// MI455X (gfx1250) — compile-verified
//
#include <hip/hip_runtime.h>

// =====================================================================
// 3-layer GCN for MI455X (gfx1250, wave32).
//   GEMMs: V_WMMA_F32_16X16X4_F32 (exact f32 matrix path), branchless
//          inner loop, compile-time K/NW strides so all W loads are
//          GLOBAL (not FLAT) with immediate offsets.
//   Edge aggregation: wave-per-edge float4 gather + global_atomic_add_f32
//     scatter (intermediates fit in 192MB L2 -> L2-bound, not HBM).
//   Dropout: exact JAX Threefry-2x32 replication.
// Workspace layout (floats), total ~61.6 MB (assumed <= ws_size):
//   dinv : 20480 | Wpad : 2048 | Hbuf : 5,120,000 | Obuf : 5,120,000
//   Abuf : 5,120,000
// =====================================================================

typedef float v2f __attribute__((ext_vector_type(2)));
typedef float v8f __attribute__((ext_vector_type(8)));

static constexpr int GCN_N  = 20000;
static constexpr int GCN_E  = 320000;
static constexpr int GCN_K0 = 512;
static constexpr int GCN_H1 = 256;
static constexpr int GCN_H2 = 128;
static constexpr int GCN_C  = 10;

// ---------------- Threefry-2x32 (exact JAX PRNG) ----------------------
__host__ __device__ inline void threefry2x32(unsigned k0, unsigned k1,
                                             unsigned x0, unsigned x1,
                                             unsigned& y0, unsigned& y1) {
  const unsigned ks0 = k0, ks1 = k1, ks2 = k0 ^ k1 ^ 0x1BD11BDAu;
  const int ra[4] = {13, 15, 26, 6};
  const int rb[4] = {17, 29, 16, 24};
  x0 += ks0; x1 += ks1;
#define TF_ROUND(rs)                                                     \
  for (int i_ = 0; i_ < 4; ++i_) {                                       \
    x0 += x1;                                                            \
    x1 = (x1 << rs[i_]) | (x1 >> (32 - rs[i_]));                         \
    x1 ^= x0;                                                            \
  }
  TF_ROUND(ra) x0 += ks1; x1 += ks2 + 1u;
  TF_ROUND(rb) x0 += ks2; x1 += ks0 + 2u;
  TF_ROUND(ra) x0 += ks0; x1 += ks1 + 3u;
  TF_ROUND(rb) x0 += ks1; x1 += ks2 + 4u;
  TF_ROUND(ra) x0 += ks2; x1 += ks0 + 5u;
#undef TF_ROUND
  y0 = x0; y1 = x1;
}

// ---------------- degree / norm ---------------------------------------
__global__ void gcn_deg_init(float* deg, int n) {
  int i = blockIdx.x * blockDim.x + threadIdx.x;
  if (i < n) deg[i] = 1.0f;  // self-loop
}

__global__ void gcn_deg_count(const int* __restrict__ dst, float* deg, int e) {
  int i = blockIdx.x * blockDim.x + threadIdx.x;
  if (i < e) atomicAdd(&deg[dst[i]], 1.0f);
}

__global__ void gcn_deg_rsqrt(float* deg, int n) {
  int i = blockIdx.x * blockDim.x + threadIdx.x;
  if (i < n) deg[i] = rsqrtf(deg[i]);
}

// ---------------- zero-pad W3 [K,10] -> Wpad [K,16] --------------------
__global__ void gcn_pad_w3(const float* __restrict__ W3, float* __restrict__ Wpad,
                           int K) {
  int i = blockIdx.x * blockDim.x + threadIdx.x;
  if (i >= K * 16) return;
  const int r = i >> 4, c = i & 15;
  Wpad[i] = (c < GCN_C) ? W3[r * GCN_C + c] : 0.0f;
}

// ---------------- WMMA f32 GEMM:  H[M,Nout] = X[M,K] @ W[K,NW] --------
// K, NW compile-time: W addressing stays anchored on the kernarg pointer
// (global addrspace inferred; immediate offsets), strides fold away.
// One wave computes a 16 x (16*NT) tile; NW is a multiple of 16*NT so
// every B load is unconditional (only the final store masks on Nout).
// A-fragment (ISA 7.12.2, 32-bit A 16x4): lanes 0-15 hold K pair {k,k+1},
// lanes 16-31 hold {k+2,k+3}, row = lane%16. B mirrors K striping,
// col = lane%16. C/D: VGPR r <-> row (r + 8*(lane/16)), col = lane%16.
template <int K, int NW, int NT>
__global__ __launch_bounds__(256) void gcn_gemm_wmma(
    const float* __restrict__ X, const float* __restrict__ W,
    float* __restrict__ H, int M, int Nout) {
  constexpr int TILEW = 16 * NT;
  const int wave = (blockIdx.x * 256 + threadIdx.x) >> 5;
  const int lane = threadIdx.x & 31;
  const int nTileN = NW / TILEW;
  const int nTileM = M >> 4;
  if (wave >= nTileM * nTileN) return;  // wave-uniform: EXEC stays all-ones
  const int tm = wave / nTileN;
  const int tn = wave % nTileN;
  const int half = lane >> 4;   // which K pair
  const int lr   = lane & 15;   // row (A) / col (B,C,D)

  const float* __restrict__ xrow = X + (long long)(tm * 16 + lr) * K;
  const int colbase = tn * TILEW + lr;

  v8f acc[NT];
#pragma unroll
  for (int t = 0; t < NT; ++t) acc[t] = (v8f)(0.0f);

#pragma unroll 4
  for (int k = 0; k < K; k += 4) {
    const int ka = k + 2 * half;                   // even -> 8B aligned
    const v2f a = *(const v2f*)(xrow + ka);        // global_load_b64
#pragma unroll
    for (int t = 0; t < NT; ++t) {
      v2f b;
      b.x = W[ka * NW + colbase + t * 16];         // row ka   (global, imm off)
      b.y = W[(ka + 1) * NW + colbase + t * 16];   // row ka+1 (global, imm off)
      acc[t] = __builtin_amdgcn_wmma_f32_16x16x4_f32(
          false, a, false, b, (short)0, acc[t], false, false);
    }
  }

#pragma unroll
  for (int t = 0; t < NT; ++t) {
    const int c = colbase + t * 16;
    if (c < Nout) {
#pragma unroll
      for (int r = 0; r < 8; ++r) {
        const int row = tm * 16 + r + 8 * half;
        H[(long long)row * Nout + c] = acc[t][r];
      }
    }
  }
}

// ---------------- self-loop init: out = h * dinv^2 ---------------------
__global__ void gcn_self_init(const float* __restrict__ h,
                              const float* __restrict__ dinv,
                              float* __restrict__ out, int n, int F) {
  long long i = (long long)blockIdx.x * blockDim.x + threadIdx.x;
  if (i >= (long long)n * F) return;
  const int node = (int)(i / F);
  const float di = dinv[node];
  out[i] = h[i] * di * di;
}

// ---------------- edge scatter: out[d] += h[s] * dinv[s]*dinv[d] -------
// Vector path: F a multiple of 128; lane handles float4 chunks.
__global__ void gcn_edge_scatter_v4(const int* __restrict__ src,
                                    const int* __restrict__ dst,
                                    const float* __restrict__ dinv,
                                    const float* __restrict__ h,
                                    float* out, int E, int F) {
  const int e = (blockIdx.x * blockDim.x + threadIdx.x) >> 5;
  const int lane = threadIdx.x & 31;
  if (e >= E) return;
  const int s = src[e];
  const int d = dst[e];
  const float nrm = dinv[s] * dinv[d];
  const float* __restrict__ hs = h + (long long)s * F;
  float* od = out + (long long)d * F;
  for (int f = lane * 4; f < F; f += 128) {
    const float4 h4 = *(const float4*)(hs + f);    // global_load_b128
    atomicAdd(&od[f + 0], h4.x * nrm);
    atomicAdd(&od[f + 1], h4.y * nrm);
    atomicAdd(&od[f + 2], h4.z * nrm);
    atomicAdd(&od[f + 3], h4.w * nrm);
  }
}

// Scalar path for small F (layer 3, F = 10).
__global__ void gcn_edge_scatter(const int* __restrict__ src,
                                 const int* __restrict__ dst,
                                 const float* __restrict__ dinv,
                                 const float* __restrict__ h,
                                 float* out, int E, int F) {
  const int e = (blockIdx.x * blockDim.x + threadIdx.x) >> 5;
  const int lane = threadIdx.x & 31;
  if (e >= E) return;
  const int s = src[e];
  const int d = dst[e];
  const float nrm = dinv[s] * dinv[d];
  const float* __restrict__ hs = h + (long long)s * F;
  float* od = out + (long long)d * F;
  for (int f = lane; f < F; f += 32)
    atomicAdd(&od[f], hs[f] * nrm);
}

// ---------------- bias + relu + dropout (exact JAX threefry) -----------
__global__ void gcn_bias_relu_drop(float* io, const float* __restrict__ bias,
                                   int n, int F, unsigned k0, unsigned k1) {
  const long long total = (long long)n * F;
  const long long halfc = total >> 1;
  const long long p = (long long)blockIdx.x * blockDim.x + threadIdx.x;
  if (p >= halfc) return;
  unsigned y0, y1;
  threefry2x32(k0, k1, (unsigned)p, (unsigned)(p + halfc), y0, y1);
  {
    const float u = __uint_as_float((y0 >> 9) | 0x3f800000u) - 1.0f;
    float v = io[p] + bias[(int)(p % F)];
    v = fmaxf(v, 0.0f);
    io[p] = (u < 0.8f) ? (v / 0.8f) : 0.0f;
  }
  {
    const long long j = p + halfc;
    const float u = __uint_as_float((y1 >> 9) | 0x3f800000u) - 1.0f;
    float v = io[j] + bias[(int)(j % F)];
    v = fmaxf(v, 0.0f);
    io[j] = (u < 0.8f) ? (v / 0.8f) : 0.0f;
  }
}

// ---------------- bias + log_softmax (C = 10) --------------------------
__global__ void gcn_bias_logsoftmax(const float* __restrict__ in,
                                    const float* __restrict__ bias,
                                    float* __restrict__ out, int n) {
  const int i = blockIdx.x * blockDim.x + threadIdx.x;
  if (i >= n) return;
  float v[GCN_C];
  float m = -__builtin_inff();
#pragma unroll
  for (int c = 0; c < GCN_C; ++c) {
    v[c] = in[i * GCN_C + c] + bias[c];
    m = fmaxf(m, v[c]);
  }
  float s = 0.0f;
#pragma unroll
  for (int c = 0; c < GCN_C; ++c) s += __expf(v[c] - m);
  const float l = m + __logf(s);
#pragma unroll
  for (int c = 0; c < GCN_C; ++c) out[i * GCN_C + c] = v[c] - l;
}

// ======================================================================
extern "C" void kernel_launch(void* const* d_in, const int* in_sizes, int n_in,
                              void* d_out, int out_size, void* d_ws, size_t ws_size,
                              hipStream_t stream) {
  (void)in_sizes; (void)n_in; (void)out_size; (void)ws_size;
  const float* x  = (const float*)d_in[0];
  const int*   ei = (const int*)d_in[1];        // [2, E] int32 (JAX x64 off)
  const float* W1 = (const float*)d_in[2];
  const float* b1 = (const float*)d_in[3];
  const float* W2 = (const float*)d_in[4];
  const float* b2 = (const float*)d_in[5];
  const float* W3 = (const float*)d_in[6];
  const float* b3 = (const float*)d_in[7];
  float* outp = (float*)d_out;

  const int N = GCN_N, E = GCN_E;
  const int* src = ei;
  const int* dst = ei + E;

  float* dinv = (float*)d_ws;
  float* Wpad = dinv + 20480;                 // 128*16 = 2048 floats
  float* Hbuf = Wpad + 2048;
  float* Obuf = Hbuf + 5120000;
  float* Abuf = Obuf + 5120000;

  // dk1, dk2 = jax.random.split(jax.random.key(1)) : key data = (0, 1)
  unsigned a0, a1, c0, c1;
  threefry2x32(0u, 1u, 0u, 2u, a0, c0);  // counter lane 0: (x0=0, x1=2)
  threefry2x32(0u, 1u, 1u, 3u, a1, c1);  // counter lane 1: (x0=1, x1=3)
  const unsigned dk1_0 = a0, dk1_1 = a1;   // first outputs
  const unsigned dk2_0 = c0, dk2_1 = c1;   // second outputs

  const int TB = 256;
  auto cdiv = [](long long a, long long b) { return (int)((a + b - 1) / b); };

  // ---- degrees: deg[i] = 1 + in-degree; dinv = rsqrt(deg) ----
  gcn_deg_init<<<cdiv(N, TB), TB, 0, stream>>>(dinv, N);
  gcn_deg_count<<<cdiv(E, TB), TB, 0, stream>>>(dst, dinv, E);
  gcn_deg_rsqrt<<<cdiv(N, TB), TB, 0, stream>>>(dinv, N);
  gcn_pad_w3<<<cdiv(GCN_H2 * 16, TB), TB, 0, stream>>>(W3, Wpad, GCN_H2);

  // ---- layer 1: [20000,512] @ [512,256] ----
  {
    const int F = GCN_H1;
    const int waves = (N / 16) * (F / 64);
    gcn_gemm_wmma<GCN_K0, GCN_H1, 4>
        <<<cdiv((long long)waves * 32, TB), TB, 0, stream>>>(x, W1, Hbuf, N, F);
    gcn_self_init<<<cdiv((long long)N * F, TB), TB, 0, stream>>>(Hbuf, dinv, Obuf, N, F);
    gcn_edge_scatter_v4<<<cdiv((long long)E * 32, TB), TB, 0, stream>>>(
        src, dst, dinv, Hbuf, Obuf, E, F);
    gcn_bias_relu_drop<<<cdiv((long long)N * F / 2, TB), TB, 0, stream>>>(
        Obuf, b1, N, F, dk1_0, dk1_1);
  }

  // ---- layer 2: [20000,256] @ [256,128] ----
  {
    const int F = GCN_H2;
    const int waves = (N / 16) * (F / 64);
    gcn_gemm_wmma<GCN_H1, GCN_H2, 4>
        <<<cdiv((long long)waves * 32, TB), TB, 0, stream>>>(Obuf, W2, Hbuf, N, F);
    gcn_self_init<<<cdiv((long long)N * F, TB), TB, 0, stream>>>(Hbuf, dinv, Abuf, N, F);
    gcn_edge_scatter_v4<<<cdiv((long long)E * 32, TB), TB, 0, stream>>>(
        src, dst, dinv, Hbuf, Abuf, E, F);
    gcn_bias_relu_drop<<<cdiv((long long)N * F / 2, TB), TB, 0, stream>>>(
        Abuf, b2, N, F, dk2_0, dk2_1);
  }

  // ---- layer 3: [20000,128] @ [128,10] (W padded to 16) + log_softmax ----
  {
    const int F = GCN_C;
    const int waves = (N / 16) * 1;  // one padded 16-col tile
    gcn_gemm_wmma<GCN_H2, 16, 1>
        <<<cdiv((long long)waves * 32, TB), TB, 0, stream>>>(Abuf, Wpad, Hbuf, N, F);
    gcn_self_init<<<cdiv((long long)N * F, TB), TB, 0, stream>>>(Hbuf, dinv, Obuf, N, F);
    gcn_edge_scatter<<<cdiv((long long)E * 32, TB), TB, 0, stream>>>(
        src, dst, dinv, Hbuf, Obuf, E, F);
    gcn_bias_logsoftmax<<<cdiv(N, TB), TB, 0, stream>>>(Obuf, b3, outp, N);
  }
}